// SGC_17257178595677
// MI455X (gfx1250) — compile-verified
//
#include <hip/hip_runtime.h>
#include <hip/hip_bf16.h>

typedef __attribute__((ext_vector_type(2))) float v2f;
typedef __attribute__((ext_vector_type(8))) float v8f;

#define N_FEAT 128
#define N_CLS  64

// ---------------------------------------------------------------------------
// deg[i] = 1.0 (self loop weight)
__global__ void sgc_deg_init(float* __restrict__ deg, int n) {
    int i = blockIdx.x * blockDim.x + threadIdx.x;
    if (i < n) deg[i] = 1.0f;
}

// deg[col[e]] += w[e]
__global__ void sgc_deg_accum(const long long* __restrict__ colIdx,
                              const float* __restrict__ ew,
                              float* __restrict__ deg, long long nnz) {
    long long e = (long long)blockIdx.x * blockDim.x + threadIdx.x;
    if (e < nnz) atomicAdd(&deg[colIdx[e]], ew[e]);
}

// deg[i] <- deg[i] > 0 ? rsqrt(max(deg,1e-38)) : 0   (in place -> d^{-1/2})
__global__ void sgc_deg_rsqrt(float* __restrict__ deg, int n) {
    int i = blockIdx.x * blockDim.x + threadIdx.x;
    if (i < n) {
        float d = deg[i];
        deg[i] = (d > 0.0f) ? rsqrtf(fmaxf(d, 1e-38f)) : 0.0f;
    }
}

// wn[e] = dis[row[e]] * w[e] * dis[col[e]]
__global__ void sgc_wnorm(const long long* __restrict__ rowIdx,
                          const long long* __restrict__ colIdx,
                          const float* __restrict__ ew,
                          const float* __restrict__ dis,
                          float* __restrict__ wn, long long nnz) {
    long long e = (long long)blockIdx.x * blockDim.x + threadIdx.x;
    if (e < nnz) wn[e] = dis[rowIdx[e]] * ew[e] * dis[colIdx[e]];
}

// ---------------------------------------------------------------------------
// h0 = x @ W^T  via V_WMMA_F32_16X16X4_F32.  One wave per 16x16 tile.
// A (16x4 f32): lane = 16*(k>=2) + m, vgpr = k&1
// B (4x16 f32): lane = 16*(k>=2) + n, vgpr = k&1
// C/D (16x16 f32): vgpr v -> M = v + 8*(lane>=16), N = lane&15
__global__ void sgc_gemm_xwt(const float* __restrict__ x,   // (n,128)
                             const float* __restrict__ w,   // (64,128)
                             float* __restrict__ h0,        // (n,64)
                             int n) {
    const int wave = (blockIdx.x * blockDim.x + threadIdx.x) >> 5;
    const int lane = threadIdx.x & 31;
    const int nTilesN = N_CLS / 16;             // 4
    const int nTilesM = n / 16;                 // n is a multiple of 16
    const int tm = wave / nTilesN;
    const int tn = wave % nTilesN;
    if (tm >= nTilesM) return;

    const int m0 = tm * 16, n0 = tn * 16;
    const int mr = lane & 15;                   // row of A / col of B
    const int kh = lane >> 4;                   // K-half selector

    const float* __restrict__ xrow = x + (size_t)(m0 + mr) * N_FEAT;
    const float* __restrict__ wrow = w + (size_t)(n0 + mr) * N_FEAT;

    v8f acc = {};
    #pragma unroll
    for (int k = 0; k < N_FEAT; k += 4) {
        v2f a, b;
        a.x = xrow[k + 2 * kh];
        a.y = xrow[k + 2 * kh + 1];
        b.x = wrow[k + 2 * kh];
        b.y = wrow[k + 2 * kh + 1];
        acc = __builtin_amdgcn_wmma_f32_16x16x4_f32(
            /*neg_a=*/false, a, /*neg_b=*/false, b,
            /*c_mod=*/(short)0, acc, /*reuse_a=*/false, /*reuse_b=*/false);
    }

    const int col = n0 + mr;
    #pragma unroll
    for (int v = 0; v < 8; ++v) {
        const int row = m0 + v + 8 * kh;
        h0[(size_t)row * N_CLS + col] = acc[v];
    }
}

// ---------------------------------------------------------------------------
// hout[i,:] = dis[i]^2 * hin[i,:]  (+ bias if useBias)  — self-loop term,
// also fully initializes hout for the edge-scatter pass.
__global__ void sgc_selfloop_init(const float* __restrict__ dis,
                                  const float* __restrict__ hin,
                                  const float* __restrict__ bias,
                                  float* __restrict__ hout,
                                  int n, int useBias) {
    long long t = (long long)blockIdx.x * blockDim.x + threadIdx.x;
    if (t >= (long long)n * 32) return;
    const int i = (int)(t >> 5);
    const int l = (int)(t & 31);
    const float d = dis[i];
    const float s = d * d;
    const float2 v = ((const float2*)(hin + (size_t)i * N_CLS))[l];
    float2 o;
    o.x = s * v.x;
    o.y = s * v.y;
    if (useBias) {
        o.x += bias[2 * l];
        o.y += bias[2 * l + 1];
    }
    ((float2*)(hout + (size_t)i * N_CLS))[l] = o;
}

// hout[col[e],:] += wn[e] * hin[row[e],:]   — one wave per edge, lane = float2.
__global__ void sgc_spmm_edges(const long long* __restrict__ rowIdx,
                               const long long* __restrict__ colIdx,
                               const float* __restrict__ wn,
                               const float* __restrict__ hin,
                               float* __restrict__ hout, long long nnz) {
    const int lane = threadIdx.x & 31;
    long long wave = ((long long)blockIdx.x * blockDim.x + threadIdx.x) >> 5;
    const long long stride = ((long long)gridDim.x * blockDim.x) >> 5;
    for (long long e = wave; e < nnz; e += stride) {
        const long long r = rowIdx[e];
        const long long c = colIdx[e];
        const float w = wn[e];
        // Prefetch next edge's source row into cache (global_prefetch_b8).
        if (e + stride < nnz) {
            const long long rn = rowIdx[e + stride];
            __builtin_prefetch(hin + rn * N_CLS + 2 * lane, 0, 1);
        }
        const float2 v = ((const float2*)(hin + r * N_CLS))[lane];
        float* dst = hout + c * N_CLS + 2 * lane;
        atomicAdd(dst,     w * v.x);
        atomicAdd(dst + 1, w * v.y);
    }
}

// ---------------------------------------------------------------------------
extern "C" void kernel_launch(void* const* d_in, const int* in_sizes, int n_in,
                              void* d_out, int out_size, void* d_ws, size_t ws_size,
                              hipStream_t stream) {
    const float*     x   = (const float*)d_in[0];       // (n, 128)
    const long long* ei  = (const long long*)d_in[1];   // (2, nnz) row-major
    const float*     ew  = (const float*)d_in[2];       // (nnz,)
    const float*     W   = (const float*)d_in[3];       // (64, 128)
    const float*     b   = (const float*)d_in[4];       // (64,)

    const int       n   = in_sizes[0] / N_FEAT;
    const long long nnz = in_sizes[2];
    const long long* rowIdx = ei;
    const long long* colIdx = ei + nnz;

    // Workspace layout (floats, 256B-aligned regions):
    //   deg/dis : n        | wn : nnz | h0 : n*64 | h1 : n*64
    float* deg = (float*)d_ws;
    float* wn  = deg + (((size_t)n + 63) & ~(size_t)63);
    float* h0  = wn  + (((size_t)nnz + 63) & ~(size_t)63);
    float* h1  = h0  + (size_t)n * N_CLS;
    float* out = (float*)d_out;

    const int B = 256;
    // 1) degrees + normalization
    sgc_deg_init <<<(n + B - 1) / B, B, 0, stream>>>(deg, n);
    sgc_deg_accum<<<(int)((nnz + B - 1) / B), B, 0, stream>>>(colIdx, ew, deg, nnz);
    sgc_deg_rsqrt<<<(n + B - 1) / B, B, 0, stream>>>(deg, n);
    sgc_wnorm    <<<(int)((nnz + B - 1) / B), B, 0, stream>>>(rowIdx, colIdx, ew, deg, wn, nnz);

    // 2) dense readout FIRST (associativity: A^2 X W^T = A (A (X W^T)))
    const int tiles = (n / 16) * (N_CLS / 16);
    sgc_gemm_xwt<<<(tiles + 7) / 8, B, 0, stream>>>(x, W, h0, n);

    // 3) hop 1: h1 = A_norm @ h0
    const long long nodeThreads = (long long)n * 32;
    const int spmmGrid = 4096;  // grid-stride over edges, one wave per edge
    sgc_selfloop_init<<<(int)((nodeThreads + B - 1) / B), B, 0, stream>>>(deg, h0, b, h1, n, 0);
    sgc_spmm_edges<<<spmmGrid, B, 0, stream>>>(rowIdx, colIdx, wn, h0, h1, nnz);

    // 4) hop 2 directly into d_out, folding in the bias
    sgc_selfloop_init<<<(int)((nodeThreads + B - 1) / B), B, 0, stream>>>(deg, h1, b, out, n, 1);
    sgc_spmm_edges<<<spmmGrid, B, 0, stream>>>(rowIdx, colIdx, wn, h1, out, nnz);
}